// Model_11673721110984
// MI455X (gfx1250) — compile-verified
//
#include <hip/hip_runtime.h>
#include <math.h>
#include <stdint.h>

#define B_  8
#define P_  16384
#define M_  16
#define CI_ 16
#define CO_ 32
#define W_  17
#define KTOT (W_*CI_)   /* 272 */
#define BCHUNK 4
#define SQ_HALF 0.70710678118654752f

typedef float v2f __attribute__((ext_vector_type(2)));
typedef float v8f __attribute__((ext_vector_type(8)));

// CDNA5 async global->LDS copy (ASYNCcnt-tracked, no VGPR round trip).
// VDST = per-lane LDS byte address, VADDR = 64-bit global address (GV mode).
__device__ __forceinline__ void async_g2l_b32(unsigned lds_off, const float* gaddr) {
    asm volatile("global_load_async_to_lds_b32 %0, %1, off"
                 :: "v"(lds_off), "v"((uint64_t)(uintptr_t)gaddr)
                 : "memory");
}

__global__ __launch_bounds__(256) void pconv_wmma_kernel(
    const float* __restrict__ in_pc,       // (B,P,CI)
    const float* __restrict__ weights,     // (W, CO*CI)
    const float* __restrict__ bias,        // (P, CO)
    const float* __restrict__ w_weights,   // (P, M, W)
    const float* __restrict__ weight_res,  // (CO, CI)
    const int*   __restrict__ neighbor_id, // (P, M)
    float* __restrict__ out)               // (B,P,CO)
{
    __shared__ int   nid_s[16 * M_];            // 1 KB
    __shared__ float ww_s[16 * KTOT];           // [p][m][w]       17 KB
    __shared__ float wmat_s[CO_ * KTOT];        // [o][w*16+i]     34 KB
    __shared__ float g_s[BCHUNK * 16 * KTOT];   // [bq][p][w*16+i] 68 KB
    __shared__ float bias_s[16 * CO_];          // 2 KB
    __shared__ float wres_s[CO_ * CI_];         // [o][i] 2 KB

    const int tid = threadIdx.x;
    const int p0  = blockIdx.x * 16;

    // ---- async staging of the two large tiles (ASYNCcnt path) ----
    {
        const float* wwg = w_weights + (size_t)p0 * KTOT;   // contiguous 4352 f
        #pragma unroll
        for (int t = 0; t < 17; ++t) {                       // 4352 = 17*256
            int f = tid + t * 256;
            async_g2l_b32((unsigned)(uintptr_t)&ww_s[f], wwg + f);
        }
        // transpose-on-the-fly: wmat_s[o][w*16+i] <- weights[w][o*16+i]
        #pragma unroll
        for (int t = 0; t < 34; ++t) {                       // 8704 = 34*256
            int s   = tid + t * 256;
            int w   = s >> 9;                                // / (CO*CI)
            int rem = s & 511;
            int o   = rem >> 4;
            int i   = rem & 15;
            async_g2l_b32((unsigned)(uintptr_t)&wmat_s[o * KTOT + w * CI_ + i],
                          weights + s);
        }
    }
    // ---- small tiles via normal loads ----
    nid_s[tid] = neighbor_id[p0 * M_ + tid];
    for (int j = tid; j < 16 * CO_; j += 256)
        bias_s[j] = bias[p0 * CO_ + j];
    for (int j = tid; j < CO_ * CI_; j += 256)
        wres_s[j] = weight_res[j];

    asm volatile("s_wait_asynccnt 0x0" ::: "memory");
    __syncthreads();

    // mask ww in place: zero rows whose neighbor is the zero-pad (nid == P)
    #pragma unroll
    for (int t = 0; t < 17; ++t) {
        int f = tid + t * 256;
        int p = f / KTOT;
        int m = (f - p * KTOT) / W_;
        if (nid_s[p * M_ + m] == P_) ww_s[f] = 0.f;
    }
    __syncthreads();

    const int lane = tid & 31;
    const int wv   = tid >> 5;       // 0..7
    const int half = lane >> 4;      // K-pair select for A/B operands
    const int col  = lane & 15;      // matrix row (A) / column (B,C,D)

    for (int bc = 0; bc < B_ / BCHUNK; ++bc) {
        // ---------- stage 1: g[bq][p][w*16+i] = sum_m ww[p][m][w]*nb[b][p][m][i]
        {
            const int p = tid >> 4;
            const int i = tid & 15;
            for (int bq = 0; bq < BCHUNK; ++bq) {
                const int b = bc * BCHUNK + bq;
                const float* src = in_pc + (size_t)b * (P_ * CI_);
                float acc[W_];
                #pragma unroll
                for (int w = 0; w < W_; ++w) acc[w] = 0.f;
                #pragma unroll
                for (int m = 0; m < M_; ++m) {
                    int idx = nid_s[p * M_ + m];
                    idx = idx < P_ ? idx : (P_ - 1);        // ww already masked to 0
                    float nb = src[idx * CI_ + i];          // gathered, L2-resident
                    const float* wwp = &ww_s[p * KTOT + m * W_];
                    #pragma unroll
                    for (int w = 0; w < W_; ++w) acc[w] = fmaf(wwp[w], nb, acc[w]);
                }
                #pragma unroll
                for (int w = 0; w < W_; ++w)
                    g_s[(bq * 16 + p) * KTOT + w * CI_ + i] = acc[w];
            }
        }
        __syncthreads();
        // ---------- stage 2: per-wave 16x16 tile, D = G(16x272) * Wmat(272x16)
        {
            const int bq    = wv >> 1;
            const int b     = bc * BCHUNK + bq;
            const int obase = (wv & 1) * 16;

            v8f acc_conv = {};
            v8f acc_res  = {};

            // A: row = lane%16, K pair at 2*half ; B: col = lane%16, same K pair
            const float* gp = &g_s[(bq * 16 + col) * KTOT + 2 * half];
            const float* wp = &wmat_s[(obase + col) * KTOT + 2 * half];
            #pragma unroll
            for (int k = 0; k < KTOT; k += 4) {             // 68 WMMA k-steps
                v2f a  = *(const v2f*)(gp + k);
                v2f bb = *(const v2f*)(wp + k);
                acc_conv = __builtin_amdgcn_wmma_f32_16x16x4_f32(
                    false, a, false, bb, (short)0, acc_conv, false, false);
            }
            // residual GEMM: X(16x16) * wres^T slice, K = CI = 16
            const float* xp = in_pc + ((size_t)b * P_ + p0 + col) * CI_ + 2 * half;
            const float* rp = &wres_s[(obase + col) * CI_ + 2 * half];
            #pragma unroll
            for (int k = 0; k < CI_; k += 4) {              // 4 WMMA k-steps
                v2f a  = *(const v2f*)(xp + k);
                v2f bb = *(const v2f*)(rp + k);
                acc_res = __builtin_amdgcn_wmma_f32_16x16x4_f32(
                    false, a, false, bb, (short)0, acc_res, false, false);
            }
            // epilogue: bias + ELU + sqrt(0.5)*conv + sqrt(0.5)*res
            #pragma unroll
            for (int r = 0; r < 8; ++r) {
                int row = r + 8 * half;                 // C/D: M = r (lanes 0-15), 8+r (16-31)
                int o   = obase + col;
                float conv = acc_conv[r] + bias_s[row * CO_ + o];
                float e    = conv > 0.f ? conv : (__expf(conv) - 1.f);
                out[((size_t)b * P_ + p0 + row) * CO_ + o] =
                    SQ_HALF * e + SQ_HALF * acc_res[r];
            }
        }
        __syncthreads();
    }
}

extern "C" void kernel_launch(void* const* d_in, const int* in_sizes, int n_in,
                              void* d_out, int out_size, void* d_ws, size_t ws_size,
                              hipStream_t stream) {
    const float* in_pc       = (const float*)d_in[0];
    const float* weights     = (const float*)d_in[1];
    const float* bias        = (const float*)d_in[2];
    const float* w_weights   = (const float*)d_in[3];
    const float* weight_res  = (const float*)d_in[4];
    const int*   neighbor_id = (const int*)d_in[5];
    float* out = (float*)d_out;
    (void)in_sizes; (void)n_in; (void)out_size; (void)d_ws; (void)ws_size;

    dim3 grid(P_ / 16);   // 1024 point-tiles
    dim3 block(256);      // 8 wave32
    pconv_wmma_kernel<<<grid, block, 0, stream>>>(
        in_pc, weights, bias, w_weights, weight_res, neighbor_id, out);
}